// DummyJoiner_31009663877507
// MI455X (gfx1250) — compile-verified
//
#include <hip/hip_runtime.h>
#include <hip/hip_bf16.h>
#include <math.h>

typedef float v2f __attribute__((ext_vector_type(2)));
typedef float v4f __attribute__((ext_vector_type(4)));
typedef float v8f __attribute__((ext_vector_type(8)));

// Problem constants (fixed by reference setup_inputs)
#define BATCH 16
#define CIN   512
#define HH    100
#define WW    100
#define NPIX  (HH * WW)          // 10000
#define EOUT  256
#define KTOP  100
#define TXT   786

// LDS feat tile: 16 pixels x CIN, padded row stride for bank-conflict-free access
#define FSTRIDE 516              // 512 + 4 floats pad; rows stay 16B aligned

// Address-space-qualified pointer types for the gfx1250 async global->LDS path
typedef __attribute__((address_space(1))) int* as1_iptr;
typedef __attribute__((address_space(3))) int* as3_iptr;

// ---------------------------------------------------------------------------
// Kernel 1: 1x1 conv GEMM via fp32 WMMA + fused per-pixel mean (scores)
// grid.x = NPIX*BATCH/16 = 10000 blocks, 512 threads (16 waves)
// Each block: full E=256 rows x 16 pixels.
// Wave w computes E rows [16w, 16w+16) for the block's 16 pixels.
// ---------------------------------------------------------------------------
__global__ __launch_bounds__(512) void conv_gemm_scores_kernel(
    const float* __restrict__ feat,      // [B, CIN, H, W]
    const float* __restrict__ conv_w,    // [E, CIN]
    const float* __restrict__ conv_b,    // [E]
    float* __restrict__ src_out,         // [B, E, H, W]
    float* __restrict__ scores)          // [B, NPIX]
{
    __shared__ float sfeat[16 * FSTRIDE];   // transposed feat tile [pix][cin]
    __shared__ float wsum[16 * 32];         // per-wave partial pixel sums

    const int t    = threadIdx.x;
    const int wave = t >> 5;        // 0..15 == E-tile id
    const int lane = t & 31;
    const int half = lane >> 4;     // 0: lanes 0-15, 1: lanes 16-31
    const int ln   = lane & 15;

    const int blk = blockIdx.x;                 // 0..9999
    const int b   = blk / (NPIX / 16);          // 625 blocks per image
    const int p0  = (blk % (NPIX / 16)) * 16;   // pixel base within image

    // ---- Stage feat tile into LDS, transposed: sfeat[pix][c] ----
    // threads: t&15 = pixel, t>>4 = channel phase (0..31)
    {
        const int pix = t & 15;
        const size_t fbase = ((size_t)b * CIN) * NPIX + p0 + pix;
#if __has_builtin(__builtin_amdgcn_global_load_async_to_lds_b32)
        // CDNA5 async global->LDS scatter: per-lane LDS dest handles the
        // transpose directly, no VGPR round-trip, tracked by ASYNCcnt.
        for (int c = (t >> 4); c < CIN; c += 32) {
            __builtin_amdgcn_global_load_async_to_lds_b32(
                (as1_iptr)(feat + fbase + (size_t)c * NPIX),
                (as3_iptr)(&sfeat[pix * FSTRIDE + c]),
                /*imm offset*/0, /*cpol*/0);
        }
#if __has_builtin(__builtin_amdgcn_s_wait_asynccnt)
        __builtin_amdgcn_s_wait_asynccnt(0);
#else
        asm volatile("s_wait_asynccnt 0" ::: "memory");
#endif
#else
        for (int c = (t >> 4); c < CIN; c += 32) {
            sfeat[pix * FSTRIDE + c] = feat[fbase + (size_t)c * NPIX];
        }
#endif
    }
    __syncthreads();

    // ---- WMMA main loop ----
    // A (conv_w rows): lane holds row = wave*16 + ln; lane-half h reads K chunk k0+4h..k0+4h+3
    // B (feat): lane holds pixel ln; same K chunk assignment -> positionally consistent
    const int row0 = wave * 16;
    const float* __restrict__ wrow = conv_w + (size_t)(row0 + ln) * CIN + 4 * half;
    const float* __restrict__ frow = &sfeat[ln * FSTRIDE + 4 * half];

    v8f acc0 = {0.f, 0.f, 0.f, 0.f, 0.f, 0.f, 0.f, 0.f};
    v8f acc1 = {0.f, 0.f, 0.f, 0.f, 0.f, 0.f, 0.f, 0.f};

#pragma unroll 4
    for (int k0 = 0; k0 < CIN; k0 += 8) {
        v4f a4 = *(const v4f*)(wrow + k0);
        v4f b4 = *(const v4f*)(frow + k0);
        v2f a0 = {a4.x, a4.y};
        v2f a1 = {a4.z, a4.w};
        v2f b0 = {b4.x, b4.y};
        v2f b1 = {b4.z, b4.w};
        acc0 = __builtin_amdgcn_wmma_f32_16x16x4_f32(false, a0, false, b0,
                                                     (short)0, acc0, false, false);
        acc1 = __builtin_amdgcn_wmma_f32_16x16x4_f32(false, a1, false, b1,
                                                     (short)0, acc1, false, false);
    }

    // ---- Epilogue: add bias, store src, accumulate pixel sums ----
    // D layout: VGPR i -> E row (row0 + half*8 + i), column N = ln
    const int rbase = row0 + half * 8;
    v4f bb0 = *(const v4f*)(conv_b + rbase);
    v4f bb1 = *(const v4f*)(conv_b + rbase + 4);

    float psum = 0.f;
    const size_t obase = ((size_t)b * EOUT) * NPIX + p0 + ln;
#pragma unroll
    for (int i = 0; i < 8; ++i) {
        float bias = (i < 4) ? bb0[i] : bb1[i - 4];
        float v = acc0[i] + acc1[i] + bias;
        psum += v;
        src_out[obase + (size_t)(rbase + i) * NPIX] = v;
    }
    wsum[wave * 32 + lane] = psum;
    __syncthreads();

    // Deterministic fixed-order reduction over 16 waves x 2 halves
    if (t < 16) {
        float s = 0.f;
        for (int w = 0; w < 16; ++w) {
            s += wsum[w * 32 + t];
            s += wsum[w * 32 + 16 + t];
        }
        scores[b * NPIX + p0 + t] = s * (1.0f / (float)EOUT);
    }
}

// ---------------------------------------------------------------------------
// Kernel 2: per-batch top-K (iterative argmax, jax tie-break: lowest index)
// fused with pos_embed generation. grid.x = BATCH, 256 threads.
// ---------------------------------------------------------------------------
__global__ __launch_bounds__(256) void topk_pos_kernel(
    const float* __restrict__ scores,    // [B, NPIX]
    const float* __restrict__ coord_w,   // [E, 2]
    const float* __restrict__ coord_b,   // [E]
    const int*   __restrict__ num_topk,  // scalar
    float* __restrict__ out_pos)         // [B, K, E]
{
    __shared__ float sv[NPIX];
    __shared__ float rv[256];
    __shared__ int   ri[256];
    __shared__ int   s_widx;

    const int t = threadIdx.x;
    const int b = blockIdx.x;
    const int K = *num_topk;

    for (int i = t; i < NPIX; i += 256) sv[i] = scores[b * NPIX + i];
    __syncthreads();

    const float cwx = coord_w[2 * t + 0];
    const float cwy = coord_w[2 * t + 1];
    const float cb  = coord_b[t];

    for (int k = 0; k < K; ++k) {
        // local argmax over strided range (ascending -> keeps lowest index on tie)
        float best = -INFINITY;
        int   bi   = 0x7fffffff;
        for (int i = t; i < NPIX; i += 256) {
            float v = sv[i];
            if (v > best) { best = v; bi = i; }
        }
        rv[t] = best; ri[t] = bi;
        __syncthreads();
        for (int s = 128; s > 0; s >>= 1) {
            if (t < s) {
                float vo = rv[t + s]; int io = ri[t + s];
                if (vo > rv[t] || (vo == rv[t] && io < ri[t])) { rv[t] = vo; ri[t] = io; }
            }
            __syncthreads();
        }
        if (t == 0) {
            int w = ri[0];
            s_widx = w;
            sv[w] = -INFINITY;     // remove winner
        }
        __syncthreads();
        const int w = s_widx;
        const float xs = (float)(w % WW) * (1.0f / (float)WW);
        const float ys = (float)(w / WW) * (1.0f / (float)HH);
        // pos_embed[b,k,e] for e = t
        out_pos[((size_t)b * KTOP + k) * EOUT + t] = xs * cwx + ys * cwy + cb;
        __syncthreads();
    }
}

// ---------------------------------------------------------------------------
// Kernel 3: text projection, broadcast over batch. 1 block, 256 threads.
// ---------------------------------------------------------------------------
__global__ __launch_bounds__(256) void text_proj_kernel(
    const float* __restrict__ text_emb,  // [1, TXT]
    const float* __restrict__ text_w,    // [E, TXT]
    const float* __restrict__ text_b,    // [E]
    float* __restrict__ out_text)        // [B, 1, E]
{
    __shared__ float ste[TXT];
    const int t = threadIdx.x;
    for (int i = t; i < TXT; i += 256) ste[i] = text_emb[i];
    __syncthreads();

    const float* __restrict__ wr = text_w + (size_t)t * TXT;
    float acc = 0.f;
    for (int i = 0; i < TXT; ++i) acc = fmaf(wr[i], ste[i], acc);
    acc += text_b[t];
    for (int b = 0; b < BATCH; ++b) out_text[b * EOUT + t] = acc;
}

// ---------------------------------------------------------------------------
extern "C" void kernel_launch(void* const* d_in, const int* in_sizes, int n_in,
                              void* d_out, int out_size, void* d_ws, size_t ws_size,
                              hipStream_t stream) {
    const float* feat     = (const float*)d_in[0];
    // d_in[1] = mask (unused by reference)
    const float* text_emb = (const float*)d_in[2];
    const float* conv_w   = (const float*)d_in[3];
    const float* conv_b   = (const float*)d_in[4];
    const float* text_w   = (const float*)d_in[5];
    const float* text_b   = (const float*)d_in[6];
    const float* coord_w  = (const float*)d_in[7];
    const float* coord_b  = (const float*)d_in[8];
    const int*   num_topk = (const int*)d_in[9];

    float* out = (float*)d_out;
    const size_t o_src  = 0;
    const size_t o_pos  = (size_t)BATCH * EOUT * NPIX;               // 40,960,000
    const size_t o_text = o_pos + (size_t)BATCH * KTOP * EOUT;       // +409,600

    float* scores = (float*)d_ws;   // BATCH*NPIX floats = 640 KB

    conv_gemm_scores_kernel<<<dim3(BATCH * NPIX / 16), dim3(512), 0, stream>>>(
        feat, conv_w, conv_b, out + o_src, scores);

    topk_pos_kernel<<<dim3(BATCH), dim3(256), 0, stream>>>(
        scores, coord_w, coord_b, num_topk, out + o_pos);

    text_proj_kernel<<<dim3(1), dim3(256), 0, stream>>>(
        text_emb, text_w, text_b, out + o_text);
}